// GAT_29729763623151
// MI455X (gfx1250) — compile-verified
//
#include <hip/hip_runtime.h>
#include <hip/hip_bf16.h>
#include <math.h>

#define HEADS 4
#define HID 32
#define FDIM 128          // HEADS*HID == HEADS*OUT == IN_DIM
#define NEG_SLOPE 0.2f

typedef __attribute__((ext_vector_type(2))) float v2f;
typedef __attribute__((ext_vector_type(8))) float v8f;

// ---------------------------------------------------------------- utilities
__global__ void fill_kernel(float* __restrict__ p, float v, int n) {
  int i = blockIdx.x * blockDim.x + threadIdx.x;
  if (i < n) p[i] = v;
}

// ------------------------------------------------------------- fp32 WMMA GEMM
// C[M,128] = A[M,128] @ B[128,128], one wave per 16x16 C tile,
// K-loop over V_WMMA_F32_16X16X4_F32.
__global__ void gemm_wmma_f32(const float* __restrict__ A,
                              const float* __restrict__ B,
                              float* __restrict__ C, int M) {
  const int K = FDIM, N = FDIM;
  int wave = (int)((blockIdx.x * blockDim.x + threadIdx.x) >> 5);
  int lane = threadIdx.x & 31;
  const int ntiles = N / 16;                 // 8
  int mt = wave / ntiles;
  int nt = wave - mt * ntiles;
  int m0 = mt << 4;
  if (m0 >= M) return;                       // wave-uniform: EXEC stays all-1s

  int sub = lane & 15;                       // M (for A) / N (for B,C) sub-index
  int hi  = lane >> 4;                       // 0: lanes 0-15, 1: lanes 16-31

  // A 16x4 layout: lane L holds A[m0+L&15][k + 2*hi .. +1]
  const float* arow = A + (size_t)(m0 + sub) * K + (hi << 1);
  // B 4x16 layout: lane L holds B[k + 2*hi .. +1][n0 + L&15]
  const float* bcol = B + (size_t)(hi << 1) * N + nt * 16 + sub;

  v8f acc = {};
  for (int k = 0; k < K; k += 4) {
    v2f a, b;
    a.x = arow[k];
    a.y = arow[k + 1];
    b.x = bcol[(size_t)k * N];
    b.y = bcol[(size_t)(k + 1) * N];
    acc = __builtin_amdgcn_wmma_f32_16x16x4_f32(false, a, false, b,
                                                (short)0, acc, false, false);
  }

  // C/D layout: VGPR r -> M = m0 + 8*hi + r, N = n0 + sub
  int rbase = m0 + (hi << 3);
  float* crow = C + (size_t)rbase * N + nt * 16 + sub;
#pragma unroll
  for (int r = 0; r < 8; ++r) crow[(size_t)r * N] = acc[r];
}

// ------------------------------------------------- attention scores el/er [N,H]
__global__ void scores_kernel(const float* __restrict__ feat,
                              const float* __restrict__ al,
                              const float* __restrict__ ar,
                              float* __restrict__ el, float* __restrict__ er,
                              int n_nodes) {
  int i = blockIdx.x * blockDim.x + threadIdx.x;      // node*HEADS + h
  if (i >= n_nodes * HEADS) return;
  int node = i >> 2;
  int h    = i & 3;
  const float* f  = feat + (size_t)node * FDIM + h * HID;
  const float* a1 = al + h * HID;
  const float* a2 = ar + h * HID;
  float sl = 0.f, sr = 0.f;
#pragma unroll
  for (int d = 0; d < HID; ++d) { float v = f[d]; sl += v * a1[d]; sr += v * a2[d]; }
  el[i] = sl;
  er[i] = sr;
}

// ---------------------------------------- edge pass 1: leaky-relu + segment max
__global__ void edge_max_kernel(const int* __restrict__ src,
                                const int* __restrict__ dst,
                                const float* __restrict__ el,
                                const float* __restrict__ er,
                                float* __restrict__ ev, float* __restrict__ mx,
                                int n_edges) {
  int i = blockIdx.x * blockDim.x + threadIdx.x;      // edge*HEADS + h
  if (i >= n_edges * HEADS) return;
  int e = i >> 2, h = i & 3;
  int s = src[e], d = dst[e];
  float v = el[s * HEADS + h] + er[d * HEADS + h];
  v = v > 0.f ? v : v * NEG_SLOPE;
  ev[i] = v;
  atomicMax(&mx[d * HEADS + h], v);                   // GLOBAL_ATOMIC_MAX_NUM_F32
}

// ---------------------------------------- edge pass 2: exp + segment denominator
__global__ void edge_expsum_kernel(const int* __restrict__ dst,
                                   float* __restrict__ ev,
                                   const float* __restrict__ mx,
                                   float* __restrict__ dn, int n_edges) {
  int i = blockIdx.x * blockDim.x + threadIdx.x;
  if (i >= n_edges * HEADS) return;
  int e = i >> 2, h = i & 3;
  int d = dst[e];
  float ex = __expf(ev[i] - mx[d * HEADS + h]);
  ev[i] = ex;
  atomicAdd(&dn[d * HEADS + h], ex);
}

// ---------------------------------------- edge pass 3: normalize coefficients
__global__ void edge_norm_kernel(const int* __restrict__ dst,
                                 float* __restrict__ ev,
                                 const float* __restrict__ dn, int n_edges) {
  int i = blockIdx.x * blockDim.x + threadIdx.x;
  if (i >= n_edges * HEADS) return;
  int e = i >> 2, h = i & 3;
  ev[i] = ev[i] / dn[dst[e] * HEADS + h];
}

// ------------------- edge pass 4: weighted scatter-aggregate (coalesced rows)
__global__ void edge_aggregate_kernel(const int* __restrict__ src,
                                      const int* __restrict__ dst,
                                      const float* __restrict__ feat,
                                      const float* __restrict__ ev,
                                      float* __restrict__ accum,
                                      long long total) {
  long long i = (long long)blockIdx.x * blockDim.x + threadIdx.x;
  if (i >= total) return;
  int e = (int)(i >> 7);                              // /FDIM
  int c = (int)(i & (FDIM - 1));
  int h = c >> 5;                                     // /HID
  float a = ev[e * HEADS + h];
  atomicAdd(&accum[(size_t)dst[e] * FDIM + c],
            feat[(size_t)src[e] * FDIM + c] * a);
}

// --------------------------------------------------------- bias + ELU (layer 1)
__global__ void elu_bias_kernel(float* __restrict__ x, const float* __restrict__ b,
                                int total) {
  int i = blockIdx.x * blockDim.x + threadIdx.x;
  if (i >= total) return;
  float v = x[i] + b[i & (FDIM - 1)];
  x[i] = v > 0.f ? v : (__expf(v) - 1.f);
}

// ------------------------------------------------- bias + mean over heads (out)
__global__ void final_mean_kernel(const float* __restrict__ accum,
                                  const float* __restrict__ b,
                                  float* __restrict__ out, int n_nodes) {
  int i = blockIdx.x * blockDim.x + threadIdx.x;      // node*OUT + o
  if (i >= n_nodes * HID) return;
  int node = i >> 5, o = i & 31;
  const float* row = accum + (size_t)node * FDIM;
  float s = 0.f;
#pragma unroll
  for (int h = 0; h < HEADS; ++h) s += row[h * HID + o] + b[h * HID + o];
  out[i] = s * 0.25f;
}

// =============================================================== host driver
extern "C" void kernel_launch(void* const* d_in, const int* in_sizes, int n_in,
                              void* d_out, int out_size, void* d_ws, size_t ws_size,
                              hipStream_t stream) {
  const float* x   = (const float*)d_in[0];
  const int*   src = (const int*)d_in[1];
  const int*   dst = (const int*)d_in[2];
  const float* W1  = (const float*)d_in[3];
  const float* al1 = (const float*)d_in[4];
  const float* ar1 = (const float*)d_in[5];
  const float* b1  = (const float*)d_in[6];
  const float* W2  = (const float*)d_in[7];
  const float* al2 = (const float*)d_in[8];
  const float* ar2 = (const float*)d_in[9];
  const float* b2  = (const float*)d_in[10];
  float* out = (float*)d_out;

  int nN = in_sizes[0] / FDIM;   // 50000
  int nE = in_sizes[1];          // 800000

  // workspace layout (floats)
  float* ws    = (float*)d_ws;
  float* feat  = ws;                                 // [N,128]
  float* accum = feat  + (size_t)nN * FDIM;          // [N,128]
  float* el    = accum + (size_t)nN * FDIM;          // [N,H]
  float* er    = el    + (size_t)nN * HEADS;         // [N,H]
  float* mx    = er    + (size_t)nN * HEADS;         // [N,H]
  float* dn    = mx    + (size_t)nN * HEADS;         // [N,H]
  float* ev    = dn    + (size_t)nN * HEADS;         // [E,H]

  const int B = 256;                                 // 8 wave32s / block
  auto cdiv = [](long long a, long long b) { return (int)((a + b - 1) / b); };

  int mtiles      = (nN + 15) / 16;
  int gemm_blocks = cdiv((long long)mtiles * (FDIM / 16) * 32, B);
  int nh = nN * HEADS;
  int eh = nE * HEADS;
  long long ec = (long long)nE * FDIM;

  for (int layer = 0; layer < 2; ++layer) {
    const float* xin = (layer == 0) ? x   : accum;   // layer2 reads layer1 activations
    const float* W   = (layer == 0) ? W1  : W2;
    const float* al  = (layer == 0) ? al1 : al2;
    const float* ar  = (layer == 0) ? ar1 : ar2;

    gemm_wmma_f32<<<gemm_blocks, B, 0, stream>>>(xin, W, feat, nN);
    // stream order guarantees the GEMM consumed `accum` before we zero it
    fill_kernel<<<cdiv((long long)nN * FDIM, B), B, 0, stream>>>(accum, 0.0f, nN * FDIM);
    fill_kernel<<<cdiv(nh, B), B, 0, stream>>>(mx, -INFINITY, nh);
    fill_kernel<<<cdiv(nh, B), B, 0, stream>>>(dn, 0.0f, nh);

    scores_kernel<<<cdiv(nh, B), B, 0, stream>>>(feat, al, ar, el, er, nN);
    edge_max_kernel<<<cdiv(eh, B), B, 0, stream>>>(src, dst, el, er, ev, mx, nE);
    edge_expsum_kernel<<<cdiv(eh, B), B, 0, stream>>>(dst, ev, mx, dn, nE);
    edge_norm_kernel<<<cdiv(eh, B), B, 0, stream>>>(dst, ev, dn, nE);
    edge_aggregate_kernel<<<cdiv(ec, B), B, 0, stream>>>(src, dst, feat, ev, accum, ec);

    if (layer == 0)
      elu_bias_kernel<<<cdiv((long long)nN * FDIM, B), B, 0, stream>>>(accum, b1, nN * FDIM);
  }

  final_mean_kernel<<<cdiv((long long)nN * HID, B), B, 0, stream>>>(accum, b2, out, nN);
}